// LEIterator_4166118277268
// MI455X (gfx1250) — compile-verified
//
#include <hip/hip_runtime.h>
#include <stdint.h>

// ---------------- problem constants ----------------
#define QDIM 16
#define SDIM 256
#define KCOMB 8
#define NTUP 16          // 6 nu=2 tuples + 10 nu=3 tuples
#define N_UNITS 1328     // 48 nu2 units (t,k) + 1280 nu3 units (t,k,f)
#define N_JOBS (N_UNITS * 16)   // x 16 s-tiles = 21248
#define WAVES 2
#define LDS_STRIDE 20    // dwords per row: 16 data + 4 pad (bank-friendly, 16B aligned)
#define LDS_PER_WAVE (256 * LDS_STRIDE)

typedef __attribute__((ext_vector_type(2))) float v2f_t;
typedef __attribute__((ext_vector_type(4))) float v4f_t;
typedef __attribute__((ext_vector_type(8))) float v8f_t;

struct Tup {
  int row_base;             // first output row of this tuple's block
  int l1, l2, l3;           // l3 = -1 for nu=2 tuples
  int parent;               // nu=2 parent tuple index (nu=3 only)
  int ip[KCOMB];
  int i1[KCOMB];
  float mult[KCOMB];
};
struct Tables { Tup tp[NTUP]; };

// ---------------- device kernel ----------------
__global__ __launch_bounds__(WAVES * 32) void le_wmma_kernel(
    const float* __restrict__ p0, const float* __restrict__ p1,
    const float* __restrict__ p2, float* __restrict__ out, Tables tabs)
{
  __shared__ float lds[WAVES][LDS_PER_WAVE];   // 40 KB / block
  const int lane = threadIdx.x & 31;
  const int wv   = threadIdx.x >> 5;
  const int job  = blockIdx.x * WAVES + wv;
  if (job >= N_JOBS) return;                    // whole-wave exit only

  const int unit = job >> 4;
  const int s0   = (job & 15) << 4;             // 16-wide s tile

  const float* LE[3] = { p0, p1, p2 };

  int t, kk, ff;
  if (unit < 48) { t = unit >> 3; kk = unit & 7; ff = -1; }
  else { int u = unit - 48; t = 6 + (u >> 7); kk = (u >> 4) & 7; ff = u & 15; }
  const Tup& tp = tabs.tp[t];

  const int m = lane & 15;                      // row for A-side / col for B-side loads
  float ua[16], vb[16];                         // per-s A-vector & B-vector values
  const float scale = tp.mult[kk];              // multiplicity, applied at store (matches ref order)
  int unit_row_base;

  if (ff < 0) {
    // nu = 2 : out[f*16+q, s] = (LE[l1][ip,f,s] * LE[l2][i1,q,s]) * mult
    const float* ap = LE[tp.l1] + ((size_t)(tp.ip[kk] * QDIM + m) * SDIM + s0);
    const float* bp = LE[tp.l2] + ((size_t)(tp.i1[kk] * QDIM + m) * SDIM + s0);
#pragma unroll
    for (int j4 = 0; j4 < 16; j4 += 4) {
      v4f_t av = *(const v4f_t*)(ap + j4);
      v4f_t bv = *(const v4f_t*)(bp + j4);
#pragma unroll
      for (int e = 0; e < 4; ++e) { ua[j4 + e] = av[e]; vb[j4 + e] = bv[e]; }
    }
    unit_row_base = tp.row_base + kk * 256;
  } else {
    // nu = 3 : out[g*16+q, s] = ((c*w2) * b) * mult,  parent product c*w2 via VALU (RNE),
    // c = LE[pl1][ip2[kp], ff, s] is wave-uniform per s.
    const Tup& pt = tabs.tp[tp.parent];
    const int kp = tp.ip[kk];
    const float* cp = LE[pt.l1] + ((size_t)(pt.ip[kp] * QDIM + ff) * SDIM + s0);
    const float* wp = LE[pt.l2] + ((size_t)(pt.i1[kp] * QDIM + m) * SDIM + s0);
    const float* bp = LE[tp.l3] + ((size_t)(tp.i1[kk] * QDIM + m) * SDIM + s0);
#pragma unroll
    for (int j4 = 0; j4 < 16; j4 += 4) {
      v4f_t cv = *(const v4f_t*)(cp + j4);
      v4f_t wv4 = *(const v4f_t*)(wp + j4);
      v4f_t bv = *(const v4f_t*)(bp + j4);
#pragma unroll
      for (int e = 0; e < 4; ++e) { ua[j4 + e] = cv[e] * wv4[e]; vb[j4 + e] = bv[e]; }
    }
    unit_row_base = tp.row_base + kk * 4096 + ff * 256;
  }

  // --- rank-1 outer products on the matrix unit: one V_WMMA_F32_16X16X4_F32 per s ---
  // A: 16x4, only column K=0 nonzero (VGPR0, lanes 0-15).  B: 4x16, only row K=0 nonzero.
  // D[m,n] = u[m]*b[n]; lane holds n=lane&15, rows m = v + 8*(lane>>4) across 8 VGPRs.
  const bool lo = (lane < 16);
  const int hi = lane >> 4;
  float* L = &lds[wv][0];
#pragma unroll
  for (int j = 0; j < 16; ++j) {
    v2f_t A; A.x = lo ? ua[j] : 0.0f; A.y = 0.0f;
    v2f_t B; B.x = lo ? vb[j] : 0.0f; B.y = 0.0f;
    v8f_t C = {0.f, 0.f, 0.f, 0.f, 0.f, 0.f, 0.f, 0.f};
    v8f_t D = __builtin_amdgcn_wmma_f32_16x16x4_f32(
        /*neg_a=*/false, A, /*neg_b=*/false, B,
        /*c_mod=*/(short)0, C, /*reuse_a=*/false, /*reuse_b=*/false);
#pragma unroll
    for (int v = 0; v < 8; ++v) {
      const int p = (v + 8 * hi) * 16 + m;      // output row within the 256-row tile
      L[p * LDS_STRIDE + j] = D[v];
    }
  }

  __syncthreads();   // LDS transpose handoff (also orders ds stores vs loads)

  // --- coalesced non-temporal stores: 32 x global_store_b128 per wave (64B segments) ---
  const size_t base = (size_t)unit_row_base * SDIM + s0;
#pragma unroll
  for (int i = 0; i < 32; ++i) {
    const int flat = i * 32 + lane;             // 1024 float4 chunks: 256 rows x 4
    const int r  = flat >> 2;
    const int j4 = (flat & 3) << 2;
    v4f_t val = *(const v4f_t*)&L[r * LDS_STRIDE + j4];
    val *= scale;                               // multiplicity applied last (matches reference)
    __builtin_nontemporal_store(val, (v4f_t*)(out + base + (size_t)r * SDIM + j4));
  }
}

// ---------------- host: exact numpy default_rng(0) replication ----------------
typedef unsigned __int128 u128_t;
struct HostRng { u128_t state, inc; int has32; uint32_t buf32; };
static const u128_t PCG_MULT =
    (((u128_t)0x2360ed051fc65da4ULL) << 64) | 0x4385df649fccf645ULL;

static inline void pcg_step(HostRng& r) { r.state = r.state * PCG_MULT + r.inc; }
static inline uint64_t pcg_next64(HostRng& r) {
  pcg_step(r);
  uint64_t hi = (uint64_t)(r.state >> 64), lo = (uint64_t)r.state;
  unsigned rot = (unsigned)(r.state >> 122);
  uint64_t v = hi ^ lo;
  return (v >> rot) | (v << ((64u - rot) & 63u));
}
static inline uint32_t pcg_next32(HostRng& r) {
  if (r.has32) { r.has32 = 0; return r.buf32; }
  uint64_t n = pcg_next64(r);
  r.has32 = 1; r.buf32 = (uint32_t)(n >> 32);
  return (uint32_t)n;
}
static void seed_pcg(HostRng& r) {
  // numpy SeedSequence(0): pool=4, entropy=[0]
  const uint32_t INIT_A = 0x43b0d7e5u, MULT_A = 0x931e8875u;
  const uint32_t INIT_B = 0x8b51f9ddu, MULT_B = 0x58f38dedu;
  const uint32_t MIX_L = 0xca01f9ddu, MIX_R = 0x4973f715u;
  uint32_t pool[4]; uint32_t hc = INIT_A;
  auto hashmix = [&](uint32_t v) -> uint32_t {
    v ^= hc; hc *= MULT_A; v *= hc; v ^= v >> 16; return v; };
  auto mixf = [&](uint32_t x, uint32_t y) -> uint32_t {
    uint32_t res = x * MIX_L - y * MIX_R; res ^= res >> 16; return res; };
  for (int i = 0; i < 4; ++i) pool[i] = hashmix(i == 0 ? 0u : 0u);
  for (int s = 0; s < 4; ++s)
    for (int d = 0; d < 4; ++d)
      if (s != d) pool[d] = mixf(pool[d], hashmix(pool[s]));
  uint32_t hb = INIT_B; uint32_t w[8];
  for (int i = 0; i < 8; ++i) {
    uint32_t v = pool[i & 3];
    v ^= hb; hb *= MULT_B; v *= hb; v ^= v >> 16; w[i] = v;
  }
  uint64_t sw[4];
  for (int i = 0; i < 4; ++i) sw[i] = (uint64_t)w[2*i] | ((uint64_t)w[2*i+1] << 32);
  u128_t initstate = (((u128_t)sw[0]) << 64) | sw[1];
  u128_t initseq   = (((u128_t)sw[2]) << 64) | sw[3];
  r.state = 0; r.inc = (initseq << 1) | 1; r.has32 = 0; r.buf32 = 0;
  pcg_step(r); r.state += initstate; pcg_step(r);
}
static uint32_t lemire32(HostRng& r, uint32_t rng) {    // returns in [0, rng]
  const uint32_t rng_excl = rng + 1;
  uint64_t mm = (uint64_t)pcg_next32(r) * rng_excl;
  uint32_t leftover = (uint32_t)mm;
  if (leftover < rng_excl) {
    const uint32_t threshold = (uint32_t)((0xFFFFFFFFu - rng) % rng_excl);
    while (leftover < threshold) {
      mm = (uint64_t)pcg_next32(r) * rng_excl;
      leftover = (uint32_t)mm;
    }
  }
  return (uint32_t)(mm >> 32);
}
static void rng_integers(HostRng& r, int high, int* o) {  // rng.integers(0, high, 8)
  if (high <= 1) { for (int i = 0; i < KCOMB; ++i) o[i] = 0; return; }  // rng==0: no draws
  for (int i = 0; i < KCOMB; ++i) o[i] = (int)lemire32(r, (uint32_t)(high - 1));
}
static void rng_mult(HostRng& r, float* o) {              // (rng.random(8)+0.5).astype(f32)
  for (int i = 0; i < KCOMB; ++i) {
    double d = (double)(pcg_next64(r) >> 11) * (1.0 / 9007199254740992.0);
    o[i] = (float)(d + 0.5);
  }
}

extern "C" void kernel_launch(void* const* d_in, const int* in_sizes, int n_in,
                              void* d_out, int out_size, void* d_ws, size_t ws_size,
                              hipStream_t stream) {
  (void)in_sizes; (void)n_in; (void)out_size; (void)d_ws; (void)ws_size;
  const float* p0 = (const float*)d_in[0];   // [1,16,256]
  const float* p1 = (const float*)d_in[1];   // [3,16,256]
  const float* p2 = (const float*)d_in[2];   // [5,16,256]
  float* out = (float*)d_out;                // [339968, 256]

  HostRng r; seed_pcg(r);
  Tables tabs;
  int parent_of[3][3];
  int ti = 0, row = 0;
  for (int l1 = 0; l1 <= 2; ++l1)
    for (int l2 = l1; l2 <= 2; ++l2) {
      Tup& tp = tabs.tp[ti];
      tp.l1 = l1; tp.l2 = l2; tp.l3 = -1; tp.parent = -1;
      rng_integers(r, 2 * l1 + 1, tp.ip);
      rng_integers(r, 2 * l2 + 1, tp.i1);
      rng_mult(r, tp.mult);
      tp.row_base = row; row += KCOMB * QDIM * QDIM;          // 2048 rows
      parent_of[l1][l2] = ti; ++ti;
    }
  for (int l1 = 0; l1 <= 2; ++l1)
    for (int l2 = l1; l2 <= 2; ++l2)
      for (int l3 = l2; l3 <= 2; ++l3) {
        Tup& tp = tabs.tp[ti];
        tp.l1 = l1; tp.l2 = l2; tp.l3 = l3; tp.parent = parent_of[l1][l2];
        rng_integers(r, KCOMB, tp.ip);
        rng_integers(r, 2 * l3 + 1, tp.i1);
        rng_mult(r, tp.mult);
        tp.row_base = row; row += KCOMB * QDIM * QDIM * QDIM; // 32768 rows
        ++ti;
      }

  const dim3 grid(N_JOBS / WAVES), block(WAVES * 32);
  le_wmma_kernel<<<grid, block, 0, stream>>>(p0, p1, p2, out, tabs);
}